// TSApproximation_33767032881890
// MI455X (gfx1250) — compile-verified
//
#include <hip/hip_runtime.h>

typedef __attribute__((ext_vector_type(2))) float v2f;
typedef __attribute__((ext_vector_type(8))) float v8f;

#define WIDTH   1000
#define PPR     200        // patches per row (1000/5)
#define NPATCH  40000
#define NPIX    1000000
#define NTERM   10

// One wave32 handles one 5x5 patch.
//   D(16x16) = A(16x4) * B(4x16) + C   via V_WMMA_F32_16X16X4_F32, 5 K-chunks (K=20)
//   A rows  = output channels (3 used), from the patch's coefficients
//   B cols  = 16 pixels of the patch (two tiles: pixels 0-15 and 16-24)
//   C seed  = bias; D rows 0..2 stored to out[c*NPIX + pixel]
__global__ __launch_bounds__(256, 4)
void ts_wmma_kernel(const float* __restrict__ pix,
                    const float* __restrict__ coef,
                    const float* __restrict__ bias,
                    float* __restrict__ out) {
    const int lane = threadIdx.x & 31;
    const int wid  = threadIdx.x >> 5;
    const int p    = (blockIdx.x << 3) + wid;     // patch id, grid sized exactly

    const int n  = lane & 15;    // B: pixel-in-tile column / A: channel row
    const int hi = lane >> 4;    // K-half selector (ISA f32 A/B layout)

    // ---- A-matrix: coefficients. lane(L) -> A[L&15, 4j+2*hi+{0,1}] = coef[p, c, t=2j+hi, d]
    // rows 3..15 load clamped (garbage rows of D, never stored) to keep EXEC full.
    const int cch = n < 3 ? n : 2;
    const float* cbase = coef + (((size_t)p * 3 + cch) * NTERM + hi) * 2;
    v2f a0 = *(const v2f*)(cbase +  0);   // t = 0+hi
    v2f a1 = *(const v2f*)(cbase +  4);   // t = 2+hi
    v2f a2 = *(const v2f*)(cbase +  8);   // t = 4+hi
    v2f a3 = *(const v2f*)(cbase + 12);   // t = 6+hi
    v2f a4 = *(const v2f*)(cbase + 16);   // t = 8+hi

    // ---- pixel coordinates for the two column tiles (clamped padding for tile1 cols 9..15)
    const int prow = p / PPR, pcol = p % PPR;
    const int q0 = n;
    const int q1t = 16 + n;
    const int q1 = q1t < 25 ? q1t : 24;
    const int g0 = (5 * prow + q0 / 5) * WIDTH + 5 * pcol + q0 % 5;
    const int g1 = (5 * prow + q1 / 5) * WIDTH + 5 * pcol + q1 % 5;
    v2f xy0 = *(const v2f*)(pix + 2 * (size_t)g0);
    v2f xy1 = *(const v2f*)(pix + 2 * (size_t)g1);

    // ---- bias -> accumulator rows 0..2 (uniform per wave: scalar loads)
    const float b0 = bias[3 * p + 0];
    const float b1 = bias[3 * p + 1];
    const float b2 = bias[3 * p + 2];
    v8f acc0 = {b0, b1, b2, 0.f, 0.f, 0.f, 0.f, 0.f};
    v8f acc1 = acc0;

    // ---- running Vandermonde powers: this lane-half supplies exponents t = 2j + hi
    const float x0 = xy0[0], y0 = xy0[1], x1 = xy1[0], y1 = xy1[1];
    float xp0 = hi ? x0 : 1.0f, yp0 = hi ? y0 : 1.0f;
    float xp1 = hi ? x1 : 1.0f, yp1 = hi ? y1 : 1.0f;
    const float xs0 = x0 * x0, ys0 = y0 * y0;
    const float xs1 = x1 * x1, ys1 = y1 * y1;

    v2f a[5] = {a0, a1, a2, a3, a4};
#pragma unroll
    for (int j = 0; j < 5; ++j) {
        v2f bc0 = {xp0, yp0};
        v2f bc1 = {xp1, yp1};
        // 8 args: (neg_a, A, neg_b, B, c_mod, C, reuse_a, reuse_b)
        acc0 = __builtin_amdgcn_wmma_f32_16x16x4_f32(false, a[j], false, bc0,
                                                     (short)0, acc0, false, false);
        acc1 = __builtin_amdgcn_wmma_f32_16x16x4_f32(false, a[j], false, bc1,
                                                     (short)0, acc1, false, false);
        xp0 *= xs0; yp0 *= ys0;
        xp1 *= xs1; yp1 *= ys1;
    }

    // ---- D rows 0..2 (lanes 0..15) are the 3 channels of the tile's pixels
    if (lane < 16) {
        out[0 * NPIX + g0] = acc0[0];
        out[1 * NPIX + g0] = acc0[1];
        out[2 * NPIX + g0] = acc0[2];
        if (n < 9) {
            out[0 * NPIX + g1] = acc1[0];
            out[1 * NPIX + g1] = acc1[1];
            out[2 * NPIX + g1] = acc1[2];
        }
    }
}

extern "C" void kernel_launch(void* const* d_in, const int* in_sizes, int n_in,
                              void* d_out, int out_size, void* d_ws, size_t ws_size,
                              hipStream_t stream) {
    (void)in_sizes; (void)n_in; (void)out_size; (void)d_ws; (void)ws_size;
    const float* pix  = (const float*)d_in[0];   // [1e6, 2]
    const float* coef = (const float*)d_in[1];   // [40000, 3, 10, 2]
    const float* bias = (const float*)d_in[2];   // [40000, 3]
    float* out = (float*)d_out;                  // [3, 1e6]

    // 8 waves (8 patches) per 256-thread block; 5000 * 8 = 40000 patches exactly.
    ts_wmma_kernel<<<dim3(NPATCH / 8), dim3(256), 0, stream>>>(pix, coef, bias, out);
}